// MixupScale_24678882083441
// MI455X (gfx1250) — compile-verified
//
#include <hip/hip_runtime.h>
#include <math.h>

#define N_NODES 100000
#define N_EDGES 500000
#define IN_DIM 128
#define HID 128
#define EPS 1e-5f

typedef float v2f __attribute__((ext_vector_type(2)));
typedef float v8f __attribute__((ext_vector_type(8)));

// ---------------------------------------------------------------------------
// Kernel 1: edge histogram  cnt_src[n] = #edges with src==n (and dst likewise)
// ---------------------------------------------------------------------------
__global__ void edge_count_kernel(const int* __restrict__ src,
                                  const int* __restrict__ dst,
                                  int* __restrict__ cs, int* __restrict__ cd) {
    int e = blockIdx.x * blockDim.x + threadIdx.x;
    if (e < N_EDGES) {
        atomicAdd(&cs[src[e]], 1);
        atomicAdd(&cd[dst[e]], 1);
    }
}

// ---------------------------------------------------------------------------
// Kernel 2: y = x @ W^T + b  via V_WMMA_F32_16X16X4_F32 (full fp32 precision)
// Block = 256 threads = 8 waves. Block handles 16 rows x 128 cols of y.
// Wave w computes the 16x16 tile at columns [16w, 16w+16).
// A (16x4 f32): lane&15 = M, (lane>>4)*2 = K base, 2 consecutive K per lane.
// B (4x16 f32): lane&15 = N, same K striping (mirrored layout).
// D (16x16 f32, 8 VGPRs): vgpr j -> row j (+8 for lanes 16-31), col = lane&15.
// ---------------------------------------------------------------------------
__global__ void gemm_xWT_bias_kernel(const float* __restrict__ x,
                                     const float* __restrict__ W,
                                     const float* __restrict__ b,
                                     float* __restrict__ y) {
    const int lane = threadIdx.x & 31;
    const int wave = threadIdx.x >> 5;
    const int row0 = blockIdx.x * 16;
    const int col0 = wave * 16;

    const int mn = lane & 15;            // M for A, N for B
    const int kb = (lane >> 4) * 2;      // K sub-offset within each 4-wide step

    const float* __restrict__ xrow = x + (size_t)(row0 + mn) * IN_DIM;
    const float* __restrict__ wrow = W + (size_t)(col0 + mn) * IN_DIM;

    v8f c = {};
#pragma unroll
    for (int k0 = 0; k0 < IN_DIM; k0 += 4) {
        v2f a  = *reinterpret_cast<const v2f*>(xrow + k0 + kb);
        v2f bb = *reinterpret_cast<const v2f*>(wrow + k0 + kb);
        c = __builtin_amdgcn_wmma_f32_16x16x4_f32(
                /*neg_a=*/false, a, /*neg_b=*/false, bb,
                /*c_mod=*/(short)0, c, /*reuse_a=*/false, /*reuse_b=*/false);
    }

    const int rtop = (lane >> 4) * 8;
    const float bias = b[col0 + mn];
#pragma unroll
    for (int j = 0; j < 8; ++j) {
        y[(size_t)(row0 + rtop + j) * HID + (col0 + mn)] = c[j] + bias;
    }
}

// ---------------------------------------------------------------------------
// Kernel 3: count-weighted per-feature moments over nodes.
// acc[0..127]=sum_src, [128..255]=sumsq_src, [256..383]=sum_dst, [384..511]=sumsq_dst
// 128 threads/block: thread f owns feature f (coalesced 512B row reads).
// ---------------------------------------------------------------------------
__global__ void stats_kernel(const float* __restrict__ y,
                             const int* __restrict__ cs,
                             const int* __restrict__ cd,
                             float* __restrict__ acc) {
    const int f = threadIdx.x;
    float s1 = 0.f, q1 = 0.f, s2 = 0.f, q2 = 0.f;
    for (int n = blockIdx.x; n < N_NODES; n += gridDim.x) {
        float v  = y[(size_t)n * HID + f];
        float c1 = (float)cs[n];
        float c2 = (float)cd[n];
        s1 += c1 * v;  q1 += c1 * v * v;
        s2 += c2 * v;  q2 += c2 * v * v;
    }
    atomicAdd(&acc[f],       s1);
    atomicAdd(&acc[128 + f], q1);
    atomicAdd(&acc[256 + f], s2);
    atomicAdd(&acc[384 + f], q2);
}

// ---------------------------------------------------------------------------
// Kernel 4: fold BN affine + attention weights into per-feature coefficients
// coef1[f] = a1[f]*gamma[f]*rsqrt(var1[f]+eps), coef2 likewise, plus the
// global constant C = sum_f a1*(beta - g*r1*m1) + a2*(beta - g*r2*m2).
// Single block of 128 threads.
// ---------------------------------------------------------------------------
__global__ void finalize_kernel(const float* __restrict__ acc,
                                const float* __restrict__ gamma,
                                const float* __restrict__ beta,
                                const float* __restrict__ att_w,
                                float* __restrict__ coef1,
                                float* __restrict__ coef2,
                                float* __restrict__ Ctot) {
    __shared__ float red[128];
    const int f = threadIdx.x;
    const float invE = 1.0f / (float)N_EDGES;

    float m1 = acc[f] * invE;
    float v1 = acc[128 + f] * invE - m1 * m1;
    float r1 = rsqrtf(v1 + EPS);
    float m2 = acc[256 + f] * invE;
    float v2 = acc[384 + f] * invE - m2 * m2;
    float r2 = rsqrtf(v2 + EPS);

    float a1 = att_w[f];
    float a2 = att_w[HID + f];
    float g  = gamma[f];
    float be = beta[f];

    coef1[f] = a1 * g * r1;
    coef2[f] = a2 * g * r2;

    red[f] = a1 * (be - g * r1 * m1) + a2 * (be - g * r2 * m2);
    __syncthreads();
    for (int s = 64; s > 0; s >>= 1) {
        if (f < s) red[f] += red[f + s];
        __syncthreads();
    }
    if (f == 0) *Ctot = red[0];
}

// ---------------------------------------------------------------------------
// Kernel 5: per-node scalars s1[n] = <coef1, y[n,:]>, s2[n] = <coef2, y[n,:]>.
// One wave32 per node; lane l loads float4 (128 = 32*4), shuffle-reduce.
// ---------------------------------------------------------------------------
__global__ void node_score_kernel(const float* __restrict__ y,
                                  const float* __restrict__ coef1,
                                  const float* __restrict__ coef2,
                                  float* __restrict__ s1,
                                  float* __restrict__ s2) {
    const int gwave = (int)((blockIdx.x * blockDim.x + threadIdx.x) >> 5);
    const int lane  = threadIdx.x & 31;
    if (gwave >= N_NODES) return;

    float4 v  = reinterpret_cast<const float4*>(y + (size_t)gwave * HID)[lane];
    float4 c1 = reinterpret_cast<const float4*>(coef1)[lane];
    float4 c2 = reinterpret_cast<const float4*>(coef2)[lane];

    float d1 = v.x * c1.x + v.y * c1.y + v.z * c1.z + v.w * c1.w;
    float d2 = v.x * c2.x + v.y * c2.y + v.z * c2.z + v.w * c2.w;
#pragma unroll
    for (int off = 16; off > 0; off >>= 1) {
        d1 += __shfl_down(d1, off, 32);
        d2 += __shfl_down(d2, off, 32);
    }
    if (lane == 0) {
        s1[gwave] = d1;
        s2[gwave] = d2;
    }
}

// ---------------------------------------------------------------------------
// Kernel 6: per-edge output = sigmoid(s1[src]+s2[dst]+C) * 0.5 + 0.5
// (SCALE = 0.5 -> ratio*(1-SCALE)+SCALE). s1/s2 are 400KB each: L2-resident.
// ---------------------------------------------------------------------------
__global__ void edge_out_kernel(const int* __restrict__ src,
                                const int* __restrict__ dst,
                                const float* __restrict__ s1,
                                const float* __restrict__ s2,
                                const float* __restrict__ Ctot,
                                float* __restrict__ out) {
    int e = blockIdx.x * blockDim.x + threadIdx.x;
    if (e >= N_EDGES) return;
    float sc = s1[src[e]] + s2[dst[e]] + *Ctot;
    float r  = 1.0f / (1.0f + __expf(-sc));
    out[e] = r * 0.5f + 0.5f;
}

// ---------------------------------------------------------------------------
// Launch
// ---------------------------------------------------------------------------
extern "C" void kernel_launch(void* const* d_in, const int* in_sizes, int n_in,
                              void* d_out, int out_size, void* d_ws, size_t ws_size,
                              hipStream_t stream) {
    const float* x     = (const float*)d_in[0];   // [N_NODES, IN_DIM]
    const float* W     = (const float*)d_in[1];   // [HID, IN_DIM]
    const float* b     = (const float*)d_in[2];   // [HID]
    const float* gamma = (const float*)d_in[3];   // [HID]
    const float* beta  = (const float*)d_in[4];   // [HID]
    const float* att_w = (const float*)d_in[5];   // [1, 2*HID]
    const int*   src   = (const int*)d_in[6];     // [N_EDGES]
    const int*   dst   = (const int*)d_in[7];     // [N_EDGES]
    float*       out   = (float*)d_out;           // [N_EDGES]

    char* ws = (char*)d_ws;
    // Workspace layout (bytes):
    const size_t oY   = 0;                                   // y: N*HID*4 = 51,200,000
    const size_t oCS  = (size_t)N_NODES * HID * 4;           // cnt_src: 400,000
    const size_t oCD  = oCS + (size_t)N_NODES * 4;           // cnt_dst: 400,000
    const size_t oACC = oCD + (size_t)N_NODES * 4;           // acc: 512 floats
    const size_t oC1  = oACC + 512 * 4;                      // coef1: 128 floats
    const size_t oC2  = oC1 + 128 * 4;                       // coef2: 128 floats
    const size_t oCT  = oC2 + 128 * 4;                       // Ctot: 1 float (pad 512)
    const size_t oS1  = oCT + 512;                           // s1: N*4
    const size_t oS2  = oS1 + (size_t)N_NODES * 4;           // s2: N*4

    float* y     = (float*)(ws + oY);
    int*   cs    = (int*)  (ws + oCS);
    int*   cd    = (int*)  (ws + oCD);
    float* acc   = (float*)(ws + oACC);
    float* coef1 = (float*)(ws + oC1);
    float* coef2 = (float*)(ws + oC2);
    float* Ctot  = (float*)(ws + oCT);
    float* s1    = (float*)(ws + oS1);
    float* s2    = (float*)(ws + oS2);

    // Zero counts + accumulators (contiguous region oCS .. oACC+2048).
    hipMemsetAsync(ws + oCS, 0, (oACC + 512 * 4) - oCS, stream);

    // 1) edge histograms
    edge_count_kernel<<<(N_EDGES + 255) / 256, 256, 0, stream>>>(src, dst, cs, cd);

    // 2) y = x @ W^T + b  (WMMA f32; 100000 rows = 6250 * 16 exactly)
    gemm_xWT_bias_kernel<<<N_NODES / 16, 256, 0, stream>>>(x, W, b, y);

    // 3) weighted per-feature moments
    stats_kernel<<<256, 128, 0, stream>>>(y, cs, cd, acc);

    // 4) fold BN + attention into per-feature coefficients
    finalize_kernel<<<1, 128, 0, stream>>>(acc, gamma, beta, att_w, coef1, coef2, Ctot);

    // 5) per-node scalar scores (one wave32 per node)
    node_score_kernel<<<(N_NODES * 32 + 255) / 256, 256, 0, stream>>>(y, coef1, coef2, s1, s2);

    // 6) per-edge sigmoid mix
    edge_out_kernel<<<(N_EDGES + 255) / 256, 256, 0, stream>>>(src, dst, s1, s2, Ctot, out);
}